// Model_55697135894940
// MI455X (gfx1250) — compile-verified
//
#include <hip/hip_runtime.h>
#include <cmath>

typedef float v2f __attribute__((ext_vector_type(2)));
typedef float v8f __attribute__((ext_vector_type(8)));

#define N_NODES 100000
#define N_EDGES 1600000
#define F_IN    256
#define F_HID   128
#define F_OUT   52

// ---------------------------------------------------------------- degrees ---
__global__ __launch_bounds__(256)
void degree_kernel(const int* __restrict__ src, const int* __restrict__ dst,
                   float* __restrict__ degs, float* __restrict__ degd, int nE) {
    int i = blockIdx.x * blockDim.x + threadIdx.x;
    if (i < nE) {
        atomicAdd(&degs[src[i]], 1.0f);
        atomicAdd(&degd[dst[i]], 1.0f);
    }
}

__global__ __launch_bounds__(256)
void norm_kernel(const int* __restrict__ src, const int* __restrict__ dst,
                 const float* __restrict__ degs, const float* __restrict__ degd,
                 float* __restrict__ nrm, int nE) {
    int i = blockIdx.x * blockDim.x + threadIdx.x;
    if (i < nE) {
        float a = degs[src[i]];
        float b = degd[dst[i]];
        nrm[i] = rsqrtf(fmaxf(a, 1.0f)) * rsqrtf(fmaxf(b, 1.0f));
    }
}

// ----------------------------------------- pack W into WMMA B-fragment order
// Wp[((kk*NT + t)*32 + lane)*2 + {0,1}] = W[(4kk+kh)(+1)][t*16 + ncol]
// with kh = 2*(lane>>4), ncol = lane&15 (ISA B layout for 16x16x4 f32).
__global__ __launch_bounds__(256)
void pack_w1(const float* __restrict__ W, float* __restrict__ Wp) {
    int idx = blockIdx.x * blockDim.x + threadIdx.x;   // (F_IN/4)*8*32 = 16384
    if (idx >= (F_IN / 4) * 8 * 32) return;
    const int lane = idx & 31;
    const int t    = (idx >> 5) & 7;
    const int kk   = idx >> 8;
    const int kh   = (lane >> 4) << 1;
    const int ncol = lane & 15;
    const int k    = (kk << 2) + kh;
    float2 v;
    v.x = W[(size_t)(k)     * F_HID + t * 16 + ncol];
    v.y = W[(size_t)(k + 1) * F_HID + t * 16 + ncol];
    *(float2*)(Wp + (size_t)idx * 2) = v;
}

__global__ __launch_bounds__(256)
void pack_w2(const float* __restrict__ W, float* __restrict__ Wp) {
    int idx = blockIdx.x * blockDim.x + threadIdx.x;   // (F_HID/4)*4*32 = 4096
    if (idx >= (F_HID / 4) * 4 * 32) return;
    const int lane = idx & 31;
    const int t    = (idx >> 5) & 3;
    const int kk   = idx >> 7;
    const int kh   = (lane >> 4) << 1;
    const int ncol = lane & 15;
    const int n    = t * 16 + ncol;
    const int k    = (kk << 2) + kh;
    float2 v;                                          // zero-pad cols >= 52
    v.x = (n < F_OUT) ? W[(size_t)(k)     * F_OUT + n] : 0.0f;
    v.y = (n < F_OUT) ? W[(size_t)(k + 1) * F_OUT + n] : 0.0f;
    *(float2*)(Wp + (size_t)idx * 2) = v;
}

// ------------------------------------------------------- f32 WMMA GEMM #1 ---
// H[N_NODES x F_HID] = X[N_NODES x F_IN] @ W1; one wave per 16-row strip,
// 8 accumulator tiles; A and B fragments are single b64 loads.
__global__ __launch_bounds__(32)
void gemm1_wmma(const float* __restrict__ X, const float* __restrict__ Wp,
                float* __restrict__ H) {
    const int lane = threadIdx.x;          // 0..31, wave32
    const int m0   = blockIdx.x * 16;
    const int kh   = (lane >> 4) << 1;
    const int ncol = lane & 15;
    const float* xrow = X + (size_t)(m0 + (lane & 15)) * F_IN;
    const float* wp   = Wp + (lane << 1);

    v8f acc[8] = {};
    for (int kk = 0; kk < F_IN / 4; ++kk) {
        v2f a = *(const v2f*)(xrow + (kk << 2) + kh);
        const float* wk = wp + (kk << 9);              // kk*8 tiles*64 floats
#pragma unroll
        for (int t = 0; t < 8; ++t) {
            v2f b = *(const v2f*)(wk + (t << 6));
            acc[t] = __builtin_amdgcn_wmma_f32_16x16x4_f32(
                false, a, false, b, (short)0, acc[t], false, false);
        }
    }
    const int mbase = m0 + ((lane >> 4) << 3);
#pragma unroll
    for (int t = 0; t < 8; ++t)
#pragma unroll
        for (int i = 0; i < 8; ++i)
            H[(size_t)(mbase + i) * F_HID + t * 16 + ncol] = acc[t][i];
}

// ------------------------------------------------------- f32 WMMA GEMM #2 ---
// H[N_NODES x F_OUT] = X[N_NODES x F_HID] @ W2 (padded to 4 N-tiles).
__global__ __launch_bounds__(32)
void gemm2_wmma(const float* __restrict__ X, const float* __restrict__ Wp,
                float* __restrict__ H) {
    const int lane = threadIdx.x;
    const int m0   = blockIdx.x * 16;
    const int kh   = (lane >> 4) << 1;
    const int ncol = lane & 15;
    const float* xrow = X + (size_t)(m0 + (lane & 15)) * F_HID;
    const float* wp   = Wp + (lane << 1);

    v8f acc[4] = {};
    for (int kk = 0; kk < F_HID / 4; ++kk) {
        v2f a = *(const v2f*)(xrow + (kk << 2) + kh);
        const float* wk = wp + (kk << 8);              // kk*4 tiles*64 floats
#pragma unroll
        for (int t = 0; t < 4; ++t) {
            v2f b = *(const v2f*)(wk + (t << 6));
            acc[t] = __builtin_amdgcn_wmma_f32_16x16x4_f32(
                false, a, false, b, (short)0, acc[t], false, false);
        }
    }
    const int mbase = m0 + ((lane >> 4) << 3);
#pragma unroll
    for (int t = 0; t < 4; ++t) {
        const int n = t * 16 + ncol;
        if (n < F_OUT)
#pragma unroll
            for (int i = 0; i < 8; ++i)
                H[(size_t)(mbase + i) * F_OUT + n] = acc[t][i];
    }
}

// --------------------------------------------------- edge gather + scatter ---
// agg[dst[e]] += h[src[e]] * norm[e]; float4 per thread, atomic f32 adds.
template <int F4, int F>
__global__ __launch_bounds__(256)
void scatter_agg(const float* __restrict__ H, const int* __restrict__ src,
                 const int* __restrict__ dst, const float* __restrict__ nrm,
                 float* __restrict__ agg, int nE) {
    const int total  = nE * F4;
    const int stride = gridDim.x * blockDim.x;
    for (int idx = blockIdx.x * blockDim.x + threadIdx.x; idx < total; idx += stride) {
        const int e = idx / F4;                 // constant divisor -> cheap
        const int f = (idx - e * F4) << 2;
        const int s = src[e];
        const int d = dst[e];
        const float nm = nrm[e];
        const float4 v = *(const float4*)(H + (size_t)s * F + f);
        float* o = agg + (size_t)d * F + f;
        atomicAdd(o + 0, v.x * nm);
        atomicAdd(o + 1, v.y * nm);
        atomicAdd(o + 2, v.z * nm);
        atomicAdd(o + 3, v.w * nm);
    }
}

// ----------------------------------------------------------- bias + ReLU ----
__global__ __launch_bounds__(256)
void bias_relu(float* __restrict__ x, const float* __restrict__ b, int n) {
    int i = blockIdx.x * blockDim.x + threadIdx.x;
    if (i < n) {
        int f = i & (F_HID - 1);
        x[i] = fmaxf(x[i] + b[f], 0.0f);
    }
}

// --------------------------------------------- bias + log_softmax (52 cols) -
__global__ __launch_bounds__(256)
void bias_logsoftmax(float* __restrict__ out, const float* __restrict__ b2,
                     int nNodes) {
    const int wave = threadIdx.x >> 5;
    const int lane = threadIdx.x & 31;
    const int node = blockIdx.x * 8 + wave;
    if (node >= nNodes) return;
    float* row = out + (size_t)node * F_OUT;

    float v0 = row[lane] + b2[lane];
    const bool has1 = (lane + 32) < F_OUT;
    float v1 = has1 ? (row[lane + 32] + b2[lane + 32]) : -INFINITY;

    float m = fmaxf(v0, v1);
    for (int off = 16; off > 0; off >>= 1)
        m = fmaxf(m, __shfl_xor(m, off, 32));

    float s = __expf(v0 - m) + (has1 ? __expf(v1 - m) : 0.0f);
    for (int off = 16; off > 0; off >>= 1)
        s += __shfl_xor(s, off, 32);

    const float lse = m + __logf(s);
    row[lane] = v0 - lse;
    if (has1) row[lane + 32] = v1 - lse;
}

// ---------------------------------------------------------------- launch ----
extern "C" void kernel_launch(void* const* d_in, const int* in_sizes, int n_in,
                              void* d_out, int out_size, void* d_ws, size_t ws_size,
                              hipStream_t stream) {
    const float* features = (const float*)d_in[0];
    const int*   esrc     = (const int*)d_in[1];
    const int*   edst     = (const int*)d_in[2];
    const float* W1       = (const float*)d_in[3];
    const float* b1       = (const float*)d_in[4];
    const float* W2       = (const float*)d_in[5];
    const float* b2       = (const float*)d_in[6];
    float* out = (float*)d_out;

    // workspace: h1 | agg1 | norm | degs | degd | Wp1 | Wp2   (~110 MB)
    float* h1   = (float*)d_ws;
    float* agg1 = h1   + (size_t)N_NODES * F_HID;
    float* nrm  = agg1 + (size_t)N_NODES * F_HID;
    float* degs = nrm  + (size_t)N_EDGES;
    float* degd = degs + (size_t)N_NODES;
    float* Wp1  = degd + (size_t)N_NODES;                 // 32768 floats
    float* Wp2  = Wp1  + (size_t)(F_IN / 4) * 8 * 64;     //  8192 floats
    float* h2   = h1;   // h1 is dead after layer-1 aggregation; reuse

    // zero accumulation targets (ws/out are poisoned by the harness)
    hipMemsetAsync(degs, 0, 2 * (size_t)N_NODES * sizeof(float), stream);
    hipMemsetAsync(agg1, 0, (size_t)N_NODES * F_HID * sizeof(float), stream);
    hipMemsetAsync(out,  0, (size_t)N_NODES * F_OUT * sizeof(float), stream);

    pack_w1<<<64, 256, 0, stream>>>(W1, Wp1);
    pack_w2<<<16, 256, 0, stream>>>(W2, Wp2);

    const int EB = (N_EDGES + 255) / 256;
    degree_kernel<<<EB, 256, 0, stream>>>(esrc, edst, degs, degd, N_EDGES);
    norm_kernel<<<EB, 256, 0, stream>>>(esrc, edst, degs, degd, nrm, N_EDGES);

    gemm1_wmma<<<N_NODES / 16, 32, 0, stream>>>(features, Wp1, h1);

    scatter_agg<F_HID / 4, F_HID><<<8192, 256, 0, stream>>>(h1, esrc, edst, nrm,
                                                            agg1, N_EDGES);
    bias_relu<<<(N_NODES * F_HID + 255) / 256, 256, 0, stream>>>(
        agg1, b1, N_NODES * F_HID);

    gemm2_wmma<<<N_NODES / 16, 32, 0, stream>>>(agg1, Wp2, h2);

    scatter_agg<F_OUT / 4, F_OUT><<<8192, 256, 0, stream>>>(h2, esrc, edst, nrm,
                                                            out, N_EDGES);
    bias_logsoftmax<<<(N_NODES + 7) / 8, 256, 0, stream>>>(out, b2, N_NODES);
}